// BullsEyePredictor_15418932593225
// MI455X (gfx1250) — compile-verified
//
#include <hip/hip_runtime.h>

#define N_NODES 200000
#define N_EDGES 6400000
#define N_GRAPHS 512
#define EMBED 8
#define RLEN 20
#define FEAT 64
#define BN_EPS 1e-5f
#define W2_PITCH 68   // dword pitch for transposed W2 in LDS: conflict-free, 8B-aligned pairs

typedef __attribute__((ext_vector_type(2))) float v2f;
typedef __attribute__((ext_vector_type(8))) float v8f;

__device__ __forceinline__ void atomAddF(float* p, float v) {
    unsafeAtomicAdd(p, v);   // hardware global_atomic_add_f32
}

// ---------------------------------------------------------------- degrees
__global__ __launch_bounds__(256) void k_degree(const int* __restrict__ ei,
                                                int* __restrict__ degi) {
    int e = blockIdx.x * 256 + threadIdx.x;
    if (e < N_EDGES) atomicAdd(&degi[ei[N_EDGES + e]], 1);
}

__global__ __launch_bounds__(256) void k_dinv(const int* __restrict__ degi,
                                              float* __restrict__ dinv) {
    int i = blockIdx.x * 256 + threadIdx.x;
    if (i < N_NODES) dinv[i] = rsqrtf((float)(degi[i] + 1));  // +1 self loop
}

// ------------------------------------------------- layer1: hs = (x@W1)*dinv
// also writes self-loop init: agg = hs*dinv
__global__ __launch_bounds__(256) void k_xw1(const float* __restrict__ x,
                                             const float* __restrict__ W1,
                                             const float* __restrict__ dinv,
                                             float* __restrict__ hs,
                                             float* __restrict__ agg) {
    int idx = blockIdx.x * 256 + threadIdx.x;
    if (idx >= N_NODES * FEAT) return;
    int i = idx >> 6, f = idx & 63;
    float di = dinv[i];
    const float* xr = x + (size_t)i * 3;
    float v = xr[0] * W1[f] + xr[1] * W1[64 + f] + xr[2] * W1[128 + f];
    float s = v * di;
    hs[idx]  = s;
    agg[idx] = s * di;
}

// layer2 self-loop init: agg = hsW2*dinv  (hsW2 already carries one dinv)
__global__ __launch_bounds__(256) void k_selfinit(const float* __restrict__ hsrc,
                                                  const float* __restrict__ dinv,
                                                  float* __restrict__ agg) {
    int idx = blockIdx.x * 256 + threadIdx.x;
    if (idx >= N_NODES * FEAT) return;
    agg[idx] = hsrc[idx] * dinv[idx >> 6];
}

// ------------------------------------------------------ edge aggregation
// one wave per edge, 32 lanes x float2 = 64 feats; agg[dst] += hs[src]*dinv[dst]
__global__ __launch_bounds__(256) void k_agg(const int* __restrict__ ei,
                                             const float* __restrict__ dinv,
                                             const float* __restrict__ hs,
                                             float* __restrict__ agg) {
    int gid = blockIdx.x * 256 + threadIdx.x;
    int e = __builtin_amdgcn_readfirstlane(gid >> 5);  // wave-uniform -> SMEM loads
    if (e >= N_EDGES) return;
    int lane = gid & 31;
    if (lane == 0 && e + 4096 < N_EDGES) {
        __builtin_prefetch(&ei[e + 4096], 0, 0);            // global_prefetch_b8
        __builtin_prefetch(&ei[N_EDGES + e + 4096], 0, 0);
    }
    int s = ei[e];
    int d = ei[N_EDGES + e];
    float w = dinv[d];
    const float2 v = *(const float2*)(hs + (size_t)s * FEAT + lane * 2);
    float* p = agg + (size_t)d * FEAT + lane * 2;
    atomAddF(p,     v.x * w);
    atomAddF(p + 1, v.y * w);
}

// ------------------------------------------------------------- batch norm
__global__ __launch_bounds__(256) void k_bnstats(const float* __restrict__ h,
                                                 const float* __restrict__ bias,
                                                 float* __restrict__ stats) {
    __shared__ float ls[512];
    int tid = threadIdx.x, f = tid & 63, sub = tid >> 6;  // 4 row-groups
    float bf = bias[f];
    float s = 0.f, s2 = 0.f;
    for (int row = blockIdx.x * 4 + sub; row < N_NODES; row += gridDim.x * 4) {
        float v = h[(size_t)row * FEAT + f] + bf;
        s += v; s2 += v * v;
    }
    ls[tid] = s; ls[256 + tid] = s2;
    __syncthreads();
    if (tid < 64) {
        float ts = ls[f] + ls[64 + f] + ls[128 + f] + ls[192 + f];
        float t2 = ls[256 + f] + ls[320 + f] + ls[384 + f] + ls[448 + f];
        atomAddF(&stats[f], ts);
        atomAddF(&stats[64 + f], t2);
    }
}

__global__ void k_bnfinal(float* __restrict__ stats,
                          const float* __restrict__ gamma,
                          const float* __restrict__ beta) {
    int f = threadIdx.x;  // 64 threads
    const float n = (float)N_NODES;
    float mu  = stats[f] / n;
    float var = stats[64 + f] / n - mu * mu;
    float rs  = rsqrtf(var + BN_EPS);
    float sc  = rs * gamma[f];
    stats[128 + f] = sc;                 // scale
    stats[192 + f] = beta[f] - mu * sc;  // shift
}

__global__ __launch_bounds__(256) void k_bnapply(float* __restrict__ h,
                                                 const float* __restrict__ bias,
                                                 const float* __restrict__ stats) {
    int idx = blockIdx.x * 256 + threadIdx.x;
    if (idx >= N_NODES * FEAT) return;
    int f = idx & 63;
    float v = h[idx] + bias[f];
    h[idx] = fmaxf(v * stats[128 + f] + stats[192 + f], 0.f);
}

// --------------------------------------------- WMMA: out = (h1 @ W2) * dinv
// V_WMMA_F32_16X16X4_F32; one wave -> 16 rows x 64 cols.
// W2 staged TRANSPOSED in LDS (pitch 68 dwords): B pair (k,k+1) is one aligned
// ds_load_b64 landing directly in an even VGPR pair, conflict-free banks.
__global__ __launch_bounds__(256) void k_h1w2(const float* __restrict__ h1,
                                              const float* __restrict__ W2,
                                              const float* __restrict__ dinv,
                                              float* __restrict__ out) {
    __shared__ float w2t[64 * W2_PITCH];
    int tid = threadIdx.x;
    for (int i = tid; i < 64 * 64; i += 256) {
        int k = i >> 6, n = i & 63;
        w2t[n * W2_PITCH + k] = W2[i];   // W2[i] = W2[k][n]
    }
    __syncthreads();

    int wave = tid >> 5;
    int lane = tid & 31;
    int half = lane >> 4;   // 0: lanes 0-15, 1: lanes 16-31
    int lo   = lane & 15;
    int m0   = (blockIdx.x * 8 + wave) * 16;
    if (m0 >= N_NODES) return;   // wave-uniform; no barriers after this point

    v8f acc0 = {}, acc1 = {}, acc2 = {}, acc3 = {};
    const float* arow = h1 + (size_t)(m0 + lo) * FEAT + 2 * half;
    const float* brow = w2t + lo * W2_PITCH + 2 * half;   // + n0*W2_PITCH + kk

    for (int kk = 0; kk < 64; kk += 4) {
        // A(16x4 f32): lanes 0-15 hold K={kk,kk+1}, lanes 16-31 K={kk+2,kk+3}
        v2f a = *(const v2f*)(arow + kk);
        // B(4x16): VGPR0 = K row kk+2h, VGPR1 = kk+2h+1, N = n0+lo
        v2f b0 = *(const v2f*)(brow + kk);
        v2f b1 = *(const v2f*)(brow + kk + 16 * W2_PITCH);
        v2f b2 = *(const v2f*)(brow + kk + 32 * W2_PITCH);
        v2f b3 = *(const v2f*)(brow + kk + 48 * W2_PITCH);
        acc0 = __builtin_amdgcn_wmma_f32_16x16x4_f32(false, a, false, b0, (short)0, acc0, false, false);
        acc1 = __builtin_amdgcn_wmma_f32_16x16x4_f32(false, a, false, b1, (short)0, acc1, false, false);
        acc2 = __builtin_amdgcn_wmma_f32_16x16x4_f32(false, a, false, b2, (short)0, acc2, false, false);
        acc3 = __builtin_amdgcn_wmma_f32_16x16x4_f32(false, a, false, b3, (short)0, acc3, false, false);
    }

    // C/D layout: VGPR r -> row m0 + r + 8*half, col n0 + lo
#pragma unroll
    for (int r = 0; r < 8; ++r) {
        int row = m0 + r + 8 * half;
        float ds = dinv[row];
        float* orow = out + (size_t)row * FEAT + lo;
        orow[0]  = acc0[r] * ds;
        orow[16] = acc1[r] * ds;
        orow[32] = acc2[r] * ds;
        orow[48] = acc3[r] * ds;
    }
}

// ------------------------------------------------------------- pooling
__global__ __launch_bounds__(256) void k_pool(const float* __restrict__ h,
                                              const int* __restrict__ batch,
                                              float* __restrict__ psum,
                                              float* __restrict__ pmax,
                                              int* __restrict__ counts) {
    int idx = blockIdx.x * 256 + threadIdx.x;
    if (idx >= N_NODES * FEAT) return;
    int i = idx >> 6, f = idx & 63;
    int g = batch[i];
    float v = h[idx];
    atomAddF(&psum[g * 64 + f], v);
    atomicMax((int*)&pmax[g * 64 + f], __float_as_int(v));  // v>=0: int order == float order
    if (f == 0) atomicAdd(&counts[g], 1);
}

// ---------------------------------------------------------- recipe branch
__global__ __launch_bounds__(64) void k_recipe(const int* __restrict__ recipe,
                                               const float* __restrict__ emb,
                                               const float* __restrict__ cw,
                                               const float* __restrict__ cb,
                                               const float* __restrict__ fcw,
                                               const float* __restrict__ fcb,
                                               float* __restrict__ rvec) {
    __shared__ float remb[RLEN * EMBED];  // [l][e]
    __shared__ float rc[16 * RLEN];       // channel-major flatten
    int g = blockIdx.x, t = threadIdx.x;
    for (int i = t; i < RLEN * EMBED; i += 64) {
        int l = i >> 3, e = i & 7;
        remb[i] = emb[recipe[g * RLEN + l] * EMBED + e];
    }
    __syncthreads();
    for (int i = t; i < 16 * RLEN; i += 64) {
        int c = i / RLEN, l = i % RLEN;
        float acc = cb[c];
#pragma unroll
        for (int k = 0; k < 3; ++k) {
            int ll = l + k - 1;
            if (ll >= 0 && ll < RLEN) {
#pragma unroll
                for (int e = 0; e < EMBED; ++e)
                    acc += remb[ll * EMBED + e] * cw[(c * EMBED + e) * 3 + k];
            }
        }
        rc[i] = fmaxf(acc, 0.f);
    }
    __syncthreads();
    float acc = fcb[t];
    for (int k = 0; k < 16 * RLEN; ++k) acc += rc[k] * fcw[k * 64 + t];
    rvec[g * 64 + t] = fmaxf(acc, 0.f);
}

// ------------------------------------------------------------- MLP head
__global__ __launch_bounds__(128) void k_mlp(const float* __restrict__ psum,
                                             const float* __restrict__ pmax,
                                             const int* __restrict__ counts,
                                             const float* __restrict__ rvec,
                                             const float* __restrict__ m1w,
                                             const float* __restrict__ m1b,
                                             const float* __restrict__ m2w,
                                             const float* __restrict__ m2b,
                                             const float* __restrict__ m3w,
                                             const float* __restrict__ m3b,
                                             float* __restrict__ out) {
    __shared__ float cbuf[192];
    __shared__ float c1[128];
    __shared__ float c2[64];
    int g = blockIdx.x, t = threadIdx.x;
    if (t < 64) {
        float cnt = fmaxf((float)counts[g], 1.0f);
        cbuf[t]       = psum[g * 64 + t] / cnt;   // mean
        cbuf[128 + t] = rvec[g * 64 + t];         // rvec
    } else {
        cbuf[t] = pmax[g * 64 + (t - 64)];        // max
    }
    __syncthreads();
    float a1 = m1b[t];
    for (int k = 0; k < 192; ++k) a1 += cbuf[k] * m1w[k * 128 + t];
    c1[t] = fmaxf(a1, 0.f);
    __syncthreads();
    if (t < 64) {
        float a2 = m2b[t];
        for (int k = 0; k < 128; ++k) a2 += c1[k] * m2w[k * 64 + t];
        c2[t] = fmaxf(a2, 0.f);
    }
    __syncthreads();
    if (t < 3) {
        float a3 = m3b[t];
        for (int k = 0; k < 64; ++k) a3 += c2[k] * m3w[k * 3 + t];
        out[g * 3 + t] = a3;
    }
}

// ---------------------------------------------------------------- launch
extern "C" void kernel_launch(void* const* d_in, const int* in_sizes, int n_in,
                              void* d_out, int out_size, void* d_ws, size_t ws_size,
                              hipStream_t stream) {
    const float* x   = (const float*)d_in[0];
    const float* W1  = (const float*)d_in[1];
    const float* b1  = (const float*)d_in[2];
    const float* g1  = (const float*)d_in[3];
    const float* be1 = (const float*)d_in[4];
    const float* W2  = (const float*)d_in[5];
    const float* b2  = (const float*)d_in[6];
    const float* g2  = (const float*)d_in[7];
    const float* be2 = (const float*)d_in[8];
    const float* emb = (const float*)d_in[9];
    const float* cw  = (const float*)d_in[10];
    const float* cb  = (const float*)d_in[11];
    const float* fcw = (const float*)d_in[12];
    const float* fcb = (const float*)d_in[13];
    const float* m1w = (const float*)d_in[14];
    const float* m1b = (const float*)d_in[15];
    const float* m2w = (const float*)d_in[16];
    const float* m2b = (const float*)d_in[17];
    const float* m3w = (const float*)d_in[18];
    const float* m3b = (const float*)d_in[19];
    const int* ei    = (const int*)d_in[20];
    const int* batch = (const int*)d_in[21];
    const int* rec   = (const int*)d_in[22];

    char* w = (char*)d_ws;
    float* dinv  = (float*)w; w += (size_t)N_NODES * 4;
    int*   degi  = (int*)w;   w += (size_t)N_NODES * 4;
    float* bufA  = (float*)w; w += (size_t)N_NODES * FEAT * 4;
    float* bufB  = (float*)w; w += (size_t)N_NODES * FEAT * 4;
    float* stats = (float*)w; w += 256 * 4;
    int*   cnts  = (int*)w;   w += N_GRAPHS * 4;
    float* psum  = (float*)w; w += (size_t)N_GRAPHS * 64 * 4;
    float* pmax  = (float*)w; w += (size_t)N_GRAPHS * 64 * 4;
    float* rvec  = (float*)w; w += (size_t)N_GRAPHS * 64 * 4;

    const int nfBlocks = (N_NODES * FEAT + 255) / 256;
    const int aggBlocks = (int)(((size_t)N_EDGES * 32) / 256);

    // degrees
    hipMemsetAsync(degi, 0, (size_t)N_NODES * 4, stream);
    k_degree<<<(N_EDGES + 255) / 256, 256, 0, stream>>>(ei, degi);
    k_dinv<<<(N_NODES + 255) / 256, 256, 0, stream>>>(degi, dinv);

    // layer 1
    k_xw1<<<nfBlocks, 256, 0, stream>>>(x, W1, dinv, bufA, bufB);
    hipMemsetAsync(stats, 0, 128 * 4, stream);
    k_agg<<<aggBlocks, 256, 0, stream>>>(ei, dinv, bufA, bufB);
    k_bnstats<<<1024, 256, 0, stream>>>(bufB, b1, stats);
    k_bnfinal<<<1, 64, 0, stream>>>(stats, g1, be1);
    k_bnapply<<<nfBlocks, 256, 0, stream>>>(bufB, b1, stats);   // bufB = h1

    // layer 2
    k_h1w2<<<(N_NODES + 127) / 128, 256, 0, stream>>>(bufB, W2, dinv, bufA);
    k_selfinit<<<nfBlocks, 256, 0, stream>>>(bufA, dinv, bufB);
    hipMemsetAsync(stats, 0, 128 * 4, stream);
    k_agg<<<aggBlocks, 256, 0, stream>>>(ei, dinv, bufA, bufB);
    k_bnstats<<<1024, 256, 0, stream>>>(bufB, b2, stats);
    k_bnfinal<<<1, 64, 0, stream>>>(stats, g2, be2);
    k_bnapply<<<nfBlocks, 256, 0, stream>>>(bufB, b2, stats);   // bufB = h2

    // pooling + recipe + head
    hipMemsetAsync(cnts, 0, (size_t)(N_GRAPHS * 4 + 2 * N_GRAPHS * 64 * 4), stream);
    k_pool<<<nfBlocks, 256, 0, stream>>>(bufB, batch, psum, pmax, cnts);
    k_recipe<<<N_GRAPHS, 64, 0, stream>>>(rec, emb, cw, cb, fcw, fcb, rvec);
    k_mlp<<<N_GRAPHS, 128, 0, stream>>>(psum, pmax, cnts, rvec,
                                        m1w, m1b, m2w, m2b, m3w, m3b, (float*)d_out);
}